// LTICell_34179349741724
// MI455X (gfx1250) — compile-verified
//
#include <hip/hip_runtime.h>
#include <hip/hip_bf16.h>

// LOD=64, LSD=128, NB=8, S=8, BW=16, Z=2.0, B=32
// Key simplification: triu(tm) block (i,j) is all-zero for j<i; causal mask keeps
// j<=i; so only j==i diagonal blocks survive. Per (b,i): solve (2I - T) x = y
// with y[m<64] = err[m] + sum_k dist[k] * sum_{d>=m} tm[k,128i+m,128i+d]*err[d].
// Inverse via Newton-Schulz X <- X(2I - A X) using V_WMMA_F32_16X16X4_F32.

typedef __attribute__((ext_vector_type(2))) float v2f;
typedef __attribute__((ext_vector_type(8))) float v8f;

constexpr int STR = 132;          // padded LDS row stride (avoids bank conflicts)
constexpr int NEWTON_ITERS = 9;   // quadratic convergence from e0~0.51 -> << fp32 eps

// ---------------------------------------------------------------------------
// Kernel 1: per-batch mixture coefficients.
//   dist[b, 0:64]  = softmax(states[b].flatten() @ Wd + bd)
//   coef[b, s, 0:8] = softmax((elu(states[b,s])+1) @ Wc + bc)
// One block per batch, 128 threads.
// ---------------------------------------------------------------------------
__global__ void lti_coeffs_kernel(const float* __restrict__ states,
                                  const float* __restrict__ Wc,
                                  const float* __restrict__ bc,
                                  const float* __restrict__ Wd,
                                  const float* __restrict__ bd,
                                  float* __restrict__ dist_out,   // [B,64]
                                  float* __restrict__ coef_out)   // [B,S,8]
{
    __shared__ float s_all[1024];
    __shared__ float lg[64];   // dist logits
    __shared__ float cg[64];   // coeff logits (8 s * 8 k)

    const int b = blockIdx.x;
    const int t = threadIdx.x;

    for (int j = t; j < 1024; j += 128) s_all[j] = states[b * 1024 + j];
    __syncthreads();

    if (t < 64) {
        float acc = bd[t];
        for (int q = 0; q < 1024; ++q) acc += s_all[q] * Wd[q * 64 + t];
        lg[t] = acc;
    } else {
        const int u = t - 64;
        const int s = u >> 3, k = u & 7;
        float acc = bc[k];
        for (int q = 0; q < 128; ++q) {
            const float x = s_all[s * 128 + q];
            const float a = (x > 0.0f) ? (x + 1.0f) : __expf(x);  // elu(x)+1
            acc += a * Wc[q * 8 + k];
        }
        cg[u] = acc;
    }
    __syncthreads();

    if (t < 64) {
        float mx = lg[0];
        for (int q = 1; q < 64; ++q) mx = fmaxf(mx, lg[q]);
        float sm = 0.0f;
        for (int q = 0; q < 64; ++q) sm += __expf(lg[q] - mx);
        dist_out[b * 64 + t] = __expf(lg[t] - mx) / sm;
    } else {
        const int u = t - 64;
        const int s = u >> 3, k = u & 7;
        float mx = cg[s * 8];
        for (int q = 1; q < 8; ++q) mx = fmaxf(mx, cg[s * 8 + q]);
        float sm = 0.0f;
        for (int q = 0; q < 8; ++q) sm += __expf(cg[s * 8 + q] - mx);
        coef_out[(b * 8 + s) * 8 + k] = __expf(cg[u] - mx) / sm;
    }
}

// ---------------------------------------------------------------------------
// Kernel 2: one workgroup per (b, i). 256 threads = 8 wave32.
//   Build A = 2I - sum_k coef_k * basis_k in LDS (banded 128x128),
//   build y (64-vector), invert A via Newton-Schulz with f32 WMMA,
//   write out[b, 7-i] = A^{-1} [y; 0].
// ---------------------------------------------------------------------------
__global__ void __launch_bounds__(256) lti_solve_kernel(
    const float* __restrict__ obs,     // [B,S,64]
    const float* __restrict__ states,  // [B,S,128]
    const float* __restrict__ tm11,    // [8,64,64]
    const float* __restrict__ tm12,
    const float* __restrict__ tm21,
    const float* __restrict__ tm22,
    const float* __restrict__ tm,      // [64,1024,1024]
    const float* __restrict__ dist,    // [B,64]
    const float* __restrict__ coef,    // [B,S,8]
    float* __restrict__ out)           // [B,S,128]
{
    extern __shared__ float sm[];
    float* Am = sm;                    // 128*STR  : A
    float* Pm = sm + 128 * STR;        // 128*STR  : X (inverse iterate)
    float* Qm = sm + 2 * 128 * STR;    // 128*STR  : A*X temp
    float* yv = sm + 3 * 128 * STR;    // 64       : RHS
    float* ev = yv + 64;               // 64       : err
    float* cf = ev + 64;               // 8        : coeffs
    float* rd = cf + 8;                // 256      : reduction scratch

    const int tid = threadIdx.x;
    const int b = blockIdx.x >> 3;
    const int i = blockIdx.x & 7;

    // ---- phase 0: err, coeffs, X0 = alpha * I ----
    if (tid < 64)
        ev[tid] = obs[(b * 8 + i) * 64 + tid] - states[(b * 8 + i) * 128 + tid];
    if (tid < 8)
        cf[tid] = coef[(b * 8 + i) * 8 + tid];
    for (int e = tid; e < 128 * 128; e += 256) {
        const int m = e >> 7, n = e & 127;
        Pm[m * STR + n] = (m == n) ? 0.5f : 0.0f;
    }
    __syncthreads();

    // ---- phase 1a: A = 2I - sum_{k<8} cf[k]*basis_k (band |col-row|in[-16,16)) ----
    for (int e = tid; e < 128 * 128; e += 256) {
        const int m = e >> 7, n = e & 127;
        const int mm = m & 63, nn = n & 63;
        const int dd = nn - mm;
        float val = 0.0f;
        if (dd >= -16 && dd < 16) {
            const float* base = (m < 64) ? ((n < 64) ? tm11 : tm12)
                                         : ((n < 64) ? tm21 : tm22);
            const int off = mm * 64 + nn;
#pragma unroll
            for (int k = 0; k < 8; ++k) val += cf[k] * base[k * 4096 + off];
        }
        Am[m * STR + n] = ((m == n) ? 2.0f : 0.0f) - val;
    }

    // ---- phase 1b: y partials. thread (m=tid>>2, q=tid&3) covers d=m+q,m+q+4,.. ----
    {
        const int q = tid & 3;
        const int m = tid >> 2;                 // m in [0,64)
        const size_t rowbase = (size_t)(128 * i + m) * 1024 + (size_t)(128 * i);
        float acc = 0.0f;
        for (int k = 0; k < 64; ++k) {
            const float* trow = tm + ((size_t)k << 20) + rowbase;
            if (k < 63)
                __builtin_prefetch(tm + ((size_t)(k + 1) << 20) + rowbase, 0, 0);
            float s = 0.0f;
            for (int d = m + q; d < 64; d += 4) s += trow[d] * ev[d];
            acc += dist[b * 64 + k] * s;
        }
        rd[tid] = acc;
    }
    __syncthreads();

    if (tid < 64)
        yv[tid] = ev[tid] + rd[4 * tid] + rd[4 * tid + 1] + rd[4 * tid + 2] + rd[4 * tid + 3];

    // ---- phase 2: Newton-Schulz X <- X (2I - A X), all-f32 WMMA 16x16x4 ----
    // wave w owns output rows [16w, 16w+16). A-operand 16x4 layout: lanes 0-15
    // hold K={0,1}, lanes 16-31 hold K={2,3}. C/D: vgpr j -> row 16w+j+8*(lane>>4).
    const int w = tid >> 5;
    const int lane = tid & 31;
    const int r0 = 16 * w + (lane & 15);
    const int csel = (lane >> 4) << 1;
    const int rw = 16 * w + ((lane >> 4) << 3);

    for (int it = 0; it < NEWTON_ITERS; ++it) {
        v2f af[32];

        // --- Q = A * X ---
#pragma unroll
        for (int kc = 0; kc < 32; ++kc) {
            const int c = 4 * kc + csel;
            af[kc][0] = Am[r0 * STR + c];
            af[kc][1] = Am[r0 * STR + c + 1];
        }
        for (int nt = 0; nt < 8; ++nt) {
            const int cb = 16 * nt + (lane & 15);
            v8f acc = {};
#pragma unroll
            for (int kc = 0; kc < 32; ++kc) {
                const int rb = 4 * kc + csel;
                v2f bf;
                bf[0] = Pm[rb * STR + cb];
                bf[1] = Pm[(rb + 1) * STR + cb];
                acc = __builtin_amdgcn_wmma_f32_16x16x4_f32(
                    false, af[kc], false, bf, (short)0, acc, false, false);
            }
#pragma unroll
            for (int j = 0; j < 8; ++j) Qm[(rw + j) * STR + cb] = acc[j];
        }
        __syncthreads();

        // --- X = 2X - X*Q  (wave writes only its own 16-row panel; panel
        //     pre-loaded into registers, so in-place update is safe) ---
#pragma unroll
        for (int kc = 0; kc < 32; ++kc) {
            const int c = 4 * kc + csel;
            af[kc][0] = Pm[r0 * STR + c];
            af[kc][1] = Pm[r0 * STR + c + 1];
        }
        for (int nt = 0; nt < 8; ++nt) {
            const int cb = 16 * nt + (lane & 15);
            v8f acc = {};
#pragma unroll
            for (int kc = 0; kc < 32; ++kc) {
                const int rb = 4 * kc + csel;
                v2f bf;
                bf[0] = Qm[rb * STR + cb];
                bf[1] = Qm[(rb + 1) * STR + cb];
                acc = __builtin_amdgcn_wmma_f32_16x16x4_f32(
                    false, af[kc], false, bf, (short)0, acc, false, false);
            }
#pragma unroll
            for (int j = 0; j < 8; ++j) {
                float* p = &Pm[(rw + j) * STR + cb];
                *p = 2.0f * (*p) - acc[j];
            }
        }
        __syncthreads();
    }

    // ---- phase 3: x = X * [y; 0]; out[b, 7-i] = x (output reversed along S) ----
    if (tid < 128) {
        float x = 0.0f;
        for (int n = 0; n < 64; ++n) x += Pm[tid * STR + n] * yv[n];
        out[(b * 8 + (7 - i)) * 128 + tid] = x;
    }
}

// ---------------------------------------------------------------------------
extern "C" void kernel_launch(void* const* d_in, const int* in_sizes, int n_in,
                              void* d_out, int out_size, void* d_ws, size_t ws_size,
                              hipStream_t stream) {
    (void)in_sizes; (void)n_in; (void)out_size; (void)ws_size;

    const float* obs    = (const float*)d_in[0];
    const float* states = (const float*)d_in[1];
    const float* tm11   = (const float*)d_in[2];
    const float* tm12   = (const float*)d_in[3];
    const float* tm21   = (const float*)d_in[4];
    const float* tm22   = (const float*)d_in[5];
    const float* tm     = (const float*)d_in[6];
    const float* Wc     = (const float*)d_in[7];
    const float* bc     = (const float*)d_in[8];
    const float* Wd     = (const float*)d_in[9];
    const float* bd     = (const float*)d_in[10];
    float* out = (float*)d_out;

    float* dist = (float*)d_ws;          // [32, 64]
    float* coef = dist + 32 * 64;        // [32, 8, 8]

    lti_coeffs_kernel<<<32, 128, 0, stream>>>(states, Wc, bc, Wd, bd, dist, coef);

    const size_t lds_bytes =
        (size_t)(3 * 128 * STR + 64 + 64 + 8 + 256) * sizeof(float);  // ~203 KB < 320 KB/WGP
    lti_solve_kernel<<<256, 256, lds_bytes, stream>>>(
        obs, states, tm11, tm12, tm21, tm22, tm, dist, coef, out);
}